// MyConv2D_50483045597276
// MI455X (gfx1250) — compile-verified
//
#include <hip/hip_runtime.h>

// CDNA5 / gfx1250 implicit-GEMM 3x3 conv via V_WMMA_F32_16X16X4_F32 (exact fp32 path).
// M = C_OUT = 32, K = 288, N = N*H*W (compute-bound, AI ~72 FLOP/B).
// Round-4: register-buffered staging pipeline — global float4 loads for chunk c+1
// are issued before computing chunk c (latency hidden behind 72 WMMAs), LDS stays
// single-buffered (xs 25.3 KB + weights 36.9 KB = 62.2 KB). Weights live in LDS so
// A-fragments are ds_load_b64 with immediate offsets (low VGPR pressure).
typedef __attribute__((ext_vector_type(2))) float v2f;
typedef __attribute__((ext_vector_type(8))) float v8f;

#define HH     256
#define WW     256
#define CIN    32
#define COUT   32
#define KDIM   288                 // CIN*9, k = ci*9 + kh*3 + kw (OIHW flatten)
#define XS_CI  8                   // channels per x chunk (4 chunks)
#define XS_COLS 264                // 256 + 2 halo, padded: interior 16B-aligned
#define XS_ROWS (XS_CI * 3)        // 24 (ci_local, r) rows
#define XS_TOT (XS_ROWS * XS_COLS) // 6336 floats = 25,344 B
#define WSH_TOT (COUT * KDIM)      // 9216 floats = 36,864 B (total LDS 62,208 B)
#define NVEC   6                   // float4 staging slots per thread per chunk

// LDS column offset for local-k within an 8-channel chunk (compile-time under
// full unroll; only the lane-half select survives to runtime).
__device__ __forceinline__ constexpr int xoff(int kl) {
    return ((kl / 9) * 3 + ((kl % 9) / 3)) * XS_COLS + (kl % 3);
}

__global__ __launch_bounds__(256) void conv3x3_wmma_f32(
    const float* __restrict__ x,     // [N][CIN][H][W]
    const float* __restrict__ wgt,   // [COUT][CIN][3][3]
    const float* __restrict__ bias,  // [COUT]
    float* __restrict__ out)         // [N][COUT][H][W]
{
    __shared__ float xs[XS_TOT];     // xs[(ci*3+r)][jj'], jj' = w + 4 (jj'=3,260 = halo)
    __shared__ float wsh[WSH_TOT];   // weights, flat [co][288]

    const int n = blockIdx.x >> 8;
    const int h = blockIdx.x & 255;

    const int tid  = threadIdx.x;
    const int wave = tid >> 5;
    const int lane = tid & 31;       // wave32
    const int half = lane >> 4;
    const int col  = lane & 15;

    // ---- stage all weights into LDS (float4 copies) ----
    {
        const float4* wg4 = (const float4*)wgt;      // 2304 float4
        float4* ws4 = (float4*)wsh;
        for (int i = tid; i < WSH_TOT / 4; i += 256)
            ws4[i] = wg4[i];
    }

    // ---- precompute this thread's staging slots (chunk-invariant parts) ----
    unsigned gofs[NVEC];   // global float offset (add chunk*XS_CI*H*W per chunk)
    int      lofs[NVEC];   // LDS float offset
    bool     vld [NVEC];   // row inside image? (uniform per wave)
#pragma unroll
    for (int j = 0; j < NVEC; ++j) {
        int i   = tid + 256 * j;      // 0 .. 1535
        int row = i >> 6;             // ci_local*3 + r
        int q   = i & 63;             // float4 index within row
        int ci  = row / 3;
        int r   = row - 3 * ci;
        int gh  = h - 1 + r;
        vld[j]  = (unsigned)gh < (unsigned)HH;
        int ghc = vld[j] ? gh : 0;
        gofs[j] = (unsigned)(((n * CIN + ci) * HH + ghc) * WW + 4 * q);
        lofs[j] = row * XS_COLS + 4 + 4 * q;
    }
    // halo columns (w=-1 -> jj'=3, w=256 -> jj'=260): zero once, never overwritten
    if (tid < XS_ROWS * 2) {
        int row  = tid >> 1;
        int side = tid & 1;
        xs[row * XS_COLS + 3 + side * 257] = 0.f;
    }

    float4 buf[NVEC];
    auto load_chunk = [&](int chunk) {
#pragma unroll
        for (int j = 0; j < NVEC; ++j) {
            float4 v = make_float4(0.f, 0.f, 0.f, 0.f);
            if (vld[j])
                v = *(const float4*)(x + gofs[j] + (size_t)chunk * (XS_CI * HH * WW));
            buf[j] = v;
        }
    };
    auto store_chunk = [&]() {
#pragma unroll
        for (int j = 0; j < NVEC; ++j)
            *(float4*)&xs[lofs[j]] = buf[j];   // ds_store_b128, 16B-aligned
    };

    // N-tile columns (+3 for padded LDS layout)
    const int nbp0 = 32 * wave + col + 3;
    const int nbp1 = nbp0 + 16;

    // A-fragment LDS bases (16x4 f32 layout: row = col, K-pair offset 2*half)
    const int a0base = col * KDIM + 2 * half;          // co 0..15
    const int a1base = (16 + col) * KDIM + 2 * half;   // co 16..31

    // Accumulators (2 M-tiles x 2 N-tiles); C/D layout: VGPR v -> M = v + 8*half
    v8f acc00, acc01, acc10, acc11;
#pragma unroll
    for (int v = 0; v < 8; ++v) {
        float b0 = bias[v + 8 * half];
        float b1 = bias[16 + v + 8 * half];
        acc00[v] = b0; acc01[v] = b0;
        acc10[v] = b1; acc11[v] = b1;
    }

    // ---- software pipeline: load(c+1) in flight while computing c ----
    load_chunk(0);
    store_chunk();
    __syncthreads();

#pragma unroll
    for (int chunk = 0; chunk < 4; ++chunk) {
        if (chunk < 3)
            load_chunk(chunk + 1);    // global loads issue now, consumed after compute

        // ---- 18 K-steps of K=4 fp32 WMMA over this chunk (fully unrolled) ----
#pragma unroll
        for (int s = 0; s < 18; ++s) {
            const int kb = 4 * s;
            const int o00 = xoff(kb),     o01 = xoff(kb + 1);
            const int o10 = xoff(kb + 2), o11 = xoff(kb + 3);
            const int off0 = half ? o10 : o00;
            const int off1 = half ? o11 : o01;

            v2f a0 = *(const v2f*)&wsh[a0base + 72 * chunk + kb];
            v2f a1 = *(const v2f*)&wsh[a1base + 72 * chunk + kb];

            v2f b0, b1;
            b0[0] = xs[off0 + nbp0]; b0[1] = xs[off1 + nbp0];
            b1[0] = xs[off0 + nbp1]; b1[1] = xs[off1 + nbp1];

            acc00 = __builtin_amdgcn_wmma_f32_16x16x4_f32(
                        false, a0, false, b0, (short)0, acc00, false, false);
            acc01 = __builtin_amdgcn_wmma_f32_16x16x4_f32(
                        false, a0, false, b1, (short)0, acc01, false, false);
            acc10 = __builtin_amdgcn_wmma_f32_16x16x4_f32(
                        false, a1, false, b0, (short)0, acc10, false, false);
            acc11 = __builtin_amdgcn_wmma_f32_16x16x4_f32(
                        false, a1, false, b1, (short)0, acc11, false, false);

            if (s & 1) __builtin_amdgcn_sched_barrier(0);  // cap scheduler lookahead
        }

        if (chunk < 3) {
            __syncthreads();          // all waves done reading xs for this chunk
            store_chunk();            // write prefetched chunk into LDS
            __syncthreads();          // stores visible before next compute
        }
    }

    // ---- store (coalesced within 16-lane halves; NCHW) ----
    const int nb0 = 32 * wave + col;
    const int nb1 = nb0 + 16;
#pragma unroll
    for (int v = 0; v < 8; ++v) {
        int co0 = v + 8 * half;
        int co1 = co0 + 16;
        size_t base0 = ((size_t)(n * COUT + co0) * HH + h) * WW;
        size_t base1 = ((size_t)(n * COUT + co1) * HH + h) * WW;
        out[base0 + nb0] = acc00[v];
        out[base0 + nb1] = acc01[v];
        out[base1 + nb0] = acc10[v];
        out[base1 + nb1] = acc11[v];
    }
}

extern "C" void kernel_launch(void* const* d_in, const int* in_sizes, int n_in,
                              void* d_out, int out_size, void* d_ws, size_t ws_size,
                              hipStream_t stream) {
    (void)in_sizes; (void)n_in; (void)out_size; (void)d_ws; (void)ws_size;
    const float* x    = (const float*)d_in[0];   // [16][32][256][256]
    const float* wgt  = (const float*)d_in[1];   // [32][32][3][3]
    const float* bias = (const float*)d_in[2];   // [32]
    float* out        = (float*)d_out;           // [16][32][256][256]

    const int grid = 16 * 256;                   // one block per (image, row)
    conv3x3_wmma_f32<<<grid, 256, 0, stream>>>(x, wgt, bias, out);
}